// PtModule_73667279061057
// MI455X (gfx1250) — compile-verified
//
#include <hip/hip_runtime.h>

// Elementwise fused affine: out = (x + 5) * 2 - 1 = fma(x, 2, 9)
// 8192x8192 fp32 -> pure HBM-streaming kernel.
// MI455X plan: B128 loads/stores with non-temporal cache hints (537MB stream,
// zero reuse, don't pollute the 192MB L2), one v_fmac per element, wave32
// blocks of 256, grid-stride for full coverage.

typedef __attribute__((ext_vector_type(4))) float v4f;

__global__ __launch_bounds__(256) void ew_affine_kernel(const float* __restrict__ x,
                                                        float* __restrict__ out,
                                                        long long n) {
    const long long n4 = n >> 2;                 // number of float4 packets
    const long long tid = (long long)blockIdx.x * blockDim.x + threadIdx.x;
    const long long stride = (long long)gridDim.x * blockDim.x;

    const v4f* __restrict__ xv = (const v4f*)x;
    v4f* __restrict__ ov = (v4f*)out;

    // Main streaming loop: 128-bit NT load -> 4 fma -> 128-bit NT store.
    for (long long i = tid; i < n4; i += stride) {
        v4f v = __builtin_nontemporal_load(&xv[i]);   // global_load_b128 th:NT
        v4f r;
        r.x = __builtin_fmaf(v.x, 2.0f, 9.0f);
        r.y = __builtin_fmaf(v.y, 2.0f, 9.0f);
        r.z = __builtin_fmaf(v.z, 2.0f, 9.0f);
        r.w = __builtin_fmaf(v.w, 2.0f, 9.0f);
        __builtin_nontemporal_store(r, &ov[i]);       // global_store_b128 th:NT
    }

    // Scalar tail (n % 4 != 0). For 8192*8192 this loop body never runs,
    // but keeps the kernel correct for any size.
    const long long tail_base = n4 << 2;
    for (long long i = tail_base + tid; i < n; i += stride) {
        float v = __builtin_nontemporal_load(&x[i]);
        __builtin_nontemporal_store(__builtin_fmaf(v, 2.0f, 9.0f), &out[i]);
    }
}

extern "C" void kernel_launch(void* const* d_in, const int* in_sizes, int n_in,
                              void* d_out, int out_size, void* d_ws, size_t ws_size,
                              hipStream_t stream) {
    (void)n_in; (void)d_ws; (void)ws_size;

    const float* x = (const float*)d_in[0];
    float* out = (float*)d_out;
    const long long n = (long long)out_size;      // == in_sizes[0] == 8192*8192

    // 8192 blocks x 256 threads (wave32 -> 8 waves/block).
    // Each thread streams ~8 float4 packets (128B) -> ~2.1M in-flight-capable
    // B128 requests across the device, comfortably saturating 23.3 TB/s HBM.
    const int threads = 256;
    int blocks = 8192;
    const long long n4 = n >> 2;
    const long long needed = (n4 + threads - 1) / threads;
    if (needed < (long long)blocks) blocks = (int)(needed > 0 ? needed : 1);

    ew_affine_kernel<<<blocks, threads, 0, stream>>>(x, out, n);
}